// OneDObjectDetectionLoss_26379689132069
// MI455X (gfx1250) — compile-verified
//
#include <hip/hip_runtime.h>
#include <math.h>

// ---------------------------------------------------------------------------
// 1-D object detection loss for MI455X (gfx1250, wave32).
//   k1: per-(b,g) IoU argmax over A anchors, anchors staged in LDS via TDM
//   k2: BCE(y=0) softplus sum over all scores, WMMA-assisted wave reduction
//   k3: corrections at the <=1024 best-anchor positions (cls -s, smooth-L1)
//   k4: deterministic final reduction -> d_out[0]
// ---------------------------------------------------------------------------

typedef float v2f __attribute__((ext_vector_type(2)));
typedef float v8f __attribute__((ext_vector_type(8)));
typedef unsigned int v4u __attribute__((ext_vector_type(4)));
typedef int v8i __attribute__((ext_vector_type(8)));
typedef int v4i __attribute__((ext_vector_type(4)));

#define TILE_ANCH 2048               // anchors per LDS tile (16 KB as int2)
#define NBLK_BCE 2048
#define EPS_F 1e-8f

// Low 32 bits of a generic pointer to LDS == wave-relative LDS byte offset
// (ISA 10.2).
static __device__ inline unsigned lds_offset(const void* p) {
  return (unsigned)(unsigned long long)p;
}

// Issue a 1-D TDM copy of `nelem` int32 words from global -> LDS.
// D# built per CDNA5 ISA sec 8.3/8.4: count=1, type=2, data_size=4B,
// tensor_dim0 = tile_dim0 = nelem (<=4096), tensor_dim1=1, stride0=nelem.
// This toolchain exposes the 6-arg builtin:
//   (uint32x4 g0, int32x8 g1, int32x4 g2, int32x4 g3, int32x8 extra, i32 cpol)
static __device__ inline void tdm_load_1d(unsigned lds_addr, const void* gptr,
                                          unsigned nelem) {
  unsigned long long ga = (unsigned long long)gptr;
  v4u g0;
  g0.x = 1u;                                            // count=1, user mode
  g0.y = lds_addr;                                      // lds_addr[31:0]
  g0.z = (unsigned)ga;                                  // global_addr[31:0]
  g0.w = (unsigned)((ga >> 32) & 0x01FFFFFFu) | (2u << 30); // addr hi | type=2
  v8i g1;
  g1[0] = 0x00020000;                                   // data_size=2 (4 bytes)
  g1[1] = (int)((nelem & 0xFFFFu) << 16);               // tensor_dim0[15:0]
  g1[2] = (int)((nelem >> 16) & 0xFFFFu) | (1 << 16);   // dim0 hi | tensor_dim1=1
  g1[3] = (int)((nelem & 0xFFFFu) << 16);               // tile_dim0 = nelem
  g1[4] = 0;                                            // tile_dim1/2 unused
  g1[5] = (int)nelem;                                   // tensor_dim0_stride lo
  g1[6] = 0;
  g1[7] = 0;
  v4i z4 = {0, 0, 0, 0};                                // groups 2/3 unused (<=2D)
  v8i z8 = {0, 0, 0, 0, 0, 0, 0, 0};
  __builtin_amdgcn_tensor_load_to_lds(g0, g1, z4, z4, z8, 0);
}

// ---------------------------------------------------------------------------
// Kernel 1: argmax IoU. One block per batch b; 16 threads per GT box (G=16).
// Anchor tiles double-buffered in LDS via the Tensor Data Mover.
// ---------------------------------------------------------------------------
__global__ __launch_bounds__(256) void odl_argmax_kernel(
    const int* __restrict__ gt_bboxes,   // [B,16,2]
    const int* __restrict__ anchors,     // [A,2]
    int* __restrict__ ws_best,           // [B*16]
    int A) {
  __shared__ int s_anc[2][TILE_ANCH * 2];
  __shared__ float s_v[256];
  __shared__ int s_i[256];

  const int b = blockIdx.x;
  const int tid = threadIdx.x;
  const int g = tid >> 4;         // GT box handled by this thread (G=16)
  const int lane = tid & 15;      // 16 threads cooperate per GT box

  const float g0 = (float)gt_bboxes[(b * 16 + g) * 2 + 0];
  const float g1 = (float)gt_bboxes[(b * 16 + g) * 2 + 1];
  const float leng = g1 - g0;

  float bestv = -1.0f;            // iou >= 0, so index 0 wins on all-zero ties
  int besti = 0x7FFFFFFF;

  const int ntiles = (A + TILE_ANCH - 1) / TILE_ANCH;

  // Prologue: wave 0 DMA-loads tile 0 and waits on its TENSORcnt.
  if (tid < 32) {
    unsigned n0 = (unsigned)(min(TILE_ANCH, A) * 2);
    tdm_load_1d(lds_offset(&s_anc[0][0]), anchors, n0);
    __builtin_amdgcn_s_wait_tensorcnt(0);
  }
  __syncthreads();

  int cur = 0;
  for (int t = 0; t < ntiles; ++t) {
    const int base = t * TILE_ANCH;
    const int cnt = min(TILE_ANCH, A - base);

    // Prefetch next tile into the other buffer while we compute on this one.
    if (t + 1 < ntiles && tid < 32) {
      const int nb = t + 1;
      unsigned nn = (unsigned)(min(TILE_ANCH, A - nb * TILE_ANCH) * 2);
      tdm_load_1d(lds_offset(&s_anc[cur ^ 1][0]),
                  anchors + (size_t)nb * TILE_ANCH * 2, nn);
    }

    const int2* tp = (const int2*)&s_anc[cur][0];
    for (int k = lane; k < cnt; k += 16) {
      int2 a = tp[k];
      float a0 = (float)a.x, a1 = (float)a.y;
      float inter = fmaxf(fminf(a1, g1) - fmaxf(a0, g0), 0.0f);
      float uni = (a1 - a0) + leng - inter;
      float iou = inter / fmaxf(uni, EPS_F);
      if (iou > bestv) {           // per-thread indices increase -> first max
        bestv = iou;
        besti = base + k;
      }
    }

    if (t + 1 < ntiles) {
      if (tid < 32) __builtin_amdgcn_s_wait_tensorcnt(0);
      __syncthreads();             // tile ready + old buffer fully consumed
      cur ^= 1;
    }
  }

  s_v[tid] = bestv;
  s_i[tid] = besti;
  __syncthreads();
  if (lane == 0) {                 // merge the 16 partials for this GT box
    float bv = s_v[tid];
    int bi = s_i[tid];
    for (int j = 1; j < 16; ++j) {
      float v = s_v[tid + j];
      int i2 = s_i[tid + j];
      if (v > bv || (v == bv && i2 < bi)) { bv = v; bi = i2; }
    }
    ws_best[b * 16 + g] = bi;
  }
}

// ---------------------------------------------------------------------------
// Kernel 2: sum of BCE-with-logits terms assuming y=0 everywhere:
//   max(s,0) + log1p(exp(-|s|)).
// Wave reduction uses v_wmma_f32_16x16x4_f32 with an all-ones B matrix:
// A[m,0]=p[m], A[m,2]=p[m+16]  =>  D rows replicate p[m]+p[m+16]; summing the
// 8 D VGPRs gives half-wave sums, one shfl_xor(16) completes the wave sum.
// ---------------------------------------------------------------------------
static __device__ inline float softplus0(float x) {
  return fmaxf(x, 0.0f) + __logf(1.0f + __expf(-fabsf(x)));
}

__global__ __launch_bounds__(256) void odl_bce_kernel(
    const float* __restrict__ scores, float* __restrict__ ws_part, int N) {
  const int n4 = N >> 2;
  float acc = 0.0f;
  const float4* p = (const float4*)scores;
  for (int i = blockIdx.x * blockDim.x + threadIdx.x; i < n4;
       i += gridDim.x * blockDim.x) {
    float4 s = p[i];
    acc += softplus0(s.x) + softplus0(s.y) + softplus0(s.z) + softplus0(s.w);
  }
  if (blockIdx.x == 0 && threadIdx.x == 0) {       // scalar tail (N % 4)
    for (int i = n4 << 2; i < N; ++i) acc += softplus0(scores[i]);
  }

  // WMMA-assisted wave32 reduction.
  v2f a;  a.x = acc;  a.y = 0.0f;
  v2f bo; bo.x = 1.0f; bo.y = 1.0f;
  v8f c = {0.f, 0.f, 0.f, 0.f, 0.f, 0.f, 0.f, 0.f};
  v8f d = __builtin_amdgcn_wmma_f32_16x16x4_f32(
      false, a, false, bo, (short)0, c, false, false);
  float t = d[0] + d[1] + d[2] + d[3] + d[4] + d[5] + d[6] + d[7];
  t += __shfl_xor(t, 16, 32);                      // combine wave halves

  __shared__ float s_w[8];
  const int wid = threadIdx.x >> 5;
  if ((threadIdx.x & 31) == 0) s_w[wid] = t;
  __syncthreads();
  if (threadIdx.x == 0) {
    float s = 0.0f;
    for (int i = 0; i < 8; ++i) s += s_w[i];
    ws_part[blockIdx.x] = s;
  }
}

// ---------------------------------------------------------------------------
// Kernel 3: corrections at the best-anchor positions (one thread per (b,g)).
//   cls: subtract scores[b,best,cls] once per distinct (best,cls) in batch b
//        (scores_ready scatter-set duplicates still give 1.0).
//   reg: smooth-L1 only at best-anchor rows; duplicate best -> last g wins
//        (scatter .set last-write semantics). Zero GT coords give zero loss.
// ---------------------------------------------------------------------------
__global__ __launch_bounds__(256) void odl_corr_kernel(
    const float* __restrict__ scores,    // [B,A,3]
    const float* __restrict__ bboxes,    // [B,A,2]
    const int* __restrict__ gt_classes,  // [B*16]
    const int* __restrict__ gt_bboxes,   // [B*16*2]
    const int* __restrict__ ws_best,     // [B*16]
    float* __restrict__ ws_corr,         // [B*16]
    int A, int NP) {
  const int i = blockIdx.x * blockDim.x + threadIdx.x;
  if (i >= NP) return;
  const int b = i >> 4, g = i & 15;
  const int best = ws_best[i];
  const int cls = gt_classes[i];
  float corr = 0.0f;

  bool first = true;
  for (int gp = 0; gp < g; ++gp) {
    const int j = (b << 4) | gp;
    if (ws_best[j] == best && gt_classes[j] == cls) first = false;
  }
  if (first) corr -= scores[((size_t)b * A + best) * 3 + cls];

  bool last = true;
  for (int gp = g + 1; gp < 16; ++gp)
    if (ws_best[(b << 4) | gp] == best) last = false;
  if (last) {
    for (int j = 0; j < 2; ++j) {
      const float tcoord = (float)gt_bboxes[i * 2 + j];
      if (tcoord != 0.0f) {
        const float d = fabsf(bboxes[((size_t)b * A + best) * 2 + j] - tcoord);
        corr += (d < 1.0f) ? 0.5f * d * d : d - 0.5f;
      }
    }
  }
  ws_corr[i] = corr;
}

// ---------------------------------------------------------------------------
// Kernel 4: deterministic final reduction of partials + corrections.
// ---------------------------------------------------------------------------
__global__ __launch_bounds__(256) void odl_final_kernel(
    const float* __restrict__ ws_part, const float* __restrict__ ws_corr,
    float* __restrict__ out, int np, int nc) {
  __shared__ float sh[256];
  float acc = 0.0f;
  for (int i = threadIdx.x; i < np; i += 256) acc += ws_part[i];
  for (int i = threadIdx.x; i < nc; i += 256) acc += ws_corr[i];
  sh[threadIdx.x] = acc;
  __syncthreads();
  for (int s = 128; s > 0; s >>= 1) {
    if (threadIdx.x < s) sh[threadIdx.x] += sh[threadIdx.x + s];
    __syncthreads();
  }
  if (threadIdx.x == 0) out[0] = sh[0];
}

// ---------------------------------------------------------------------------
extern "C" void kernel_launch(void* const* d_in, const int* in_sizes, int n_in,
                              void* d_out, int out_size, void* d_ws,
                              size_t ws_size, hipStream_t stream) {
  const float* scores = (const float*)d_in[0];   // [B,A,3] f32
  const float* bboxes = (const float*)d_in[1];   // [B,A,2] f32
  const int* gt_classes = (const int*)d_in[2];   // [B,G]
  const int* gt_bboxes = (const int*)d_in[3];    // [B,G,2]
  const int* anchors = (const int*)d_in[4];      // [A,2]

  const int A = in_sizes[4] / 2;                 // 50000
  const int B = in_sizes[0] / (A * 3);           // 64
  const int G = in_sizes[2] / B;                 // 16 (kernel assumes 16)
  const int N = in_sizes[0];                     // B*A*3
  const int NP = B * G;                          // 1024 pairs

  int* ws_best = (int*)d_ws;                                   // NP ints
  float* ws_part = (float*)((char*)d_ws + 4096);               // NBLK_BCE f32
  float* ws_corr = (float*)((char*)d_ws + 4096 + NBLK_BCE * 4);// NP f32

  odl_argmax_kernel<<<B, 256, 0, stream>>>(gt_bboxes, anchors, ws_best, A);
  odl_bce_kernel<<<NBLK_BCE, 256, 0, stream>>>(scores, ws_part, N);
  odl_corr_kernel<<<(NP + 255) / 256, 256, 0, stream>>>(
      scores, bboxes, gt_classes, gt_bboxes, ws_best, ws_corr, A, NP);
  odl_final_kernel<<<1, 256, 0, stream>>>(ws_part, ws_corr, (float*)d_out,
                                          NBLK_BCE, NP);
}